// MemoryBlock_59983513256537
// MI455X (gfx1250) — compile-verified
//
#include <hip/hip_runtime.h>

typedef float v8f __attribute__((ext_vector_type(8)));
typedef float v4f __attribute__((ext_vector_type(4)));
typedef float v2f __attribute__((ext_vector_type(2)));

#define B_    16
#define N_    8192
#define FEAT_ 256
#define MEM_  128
#define U_    512

// ---------------------------------------------------------------------------
// Kernel 0: zero the S accumulator (16 x 256 floats) in workspace.
// ---------------------------------------------------------------------------
__global__ void zero_S_kernel(float* __restrict__ S) {
    int i = blockIdx.x * blockDim.x + threadIdx.x;
    if (i < B_ * FEAT_) S[i] = 0.0f;
}

// ---------------------------------------------------------------------------
// Kernel 1: S[b,f] = sum_n X[b,n,f].  Memory-bound streaming pass over X
// (134 MB, read once -> non-temporal loads). Grid: 16 batches x 64 slabs of
// 128 rows. 256 threads: 4 row-groups x 64 float4 lanes across FEAT=256.
// ---------------------------------------------------------------------------
__global__ __launch_bounds__(256) void reduce_X_kernel(const float* __restrict__ X,
                                                       float* __restrict__ S) {
    __shared__ v4f red[256];
    const int bid  = blockIdx.x;
    const int b    = bid >> 6;      // batch
    const int slab = bid & 63;      // 128-row slab within batch
    const int t    = threadIdx.x;
    const int g    = t >> 6;        // row group 0..3
    const int fq   = t & 63;        // float4 index within the 256-wide feature row

    const v4f* Xp = reinterpret_cast<const v4f*>(
        X + ((size_t)b * N_ + (size_t)slab * 128 + g) * FEAT_) + fq;

    v4f acc = {0.0f, 0.0f, 0.0f, 0.0f};
#pragma unroll 4
    for (int i = 0; i < 32; ++i) {                 // 32 iters x 4 rows = 128 rows
        v4f v = __builtin_nontemporal_load(Xp + (size_t)i * (4 * FEAT_ / 4));
        acc += v;
    }
    red[t] = acc;
    __syncthreads();

    if (t < 64) {
        v4f s = red[t] + red[t + 64] + red[t + 128] + red[t + 192];
        float* Sp = S + b * FEAT_ + t * 4;
        atomicAdd(Sp + 0, s.x);
        atomicAdd(Sp + 1, s.y);
        atomicAdd(Sp + 2, s.z);
        atomicAdd(Sp + 3, s.w);
    }
}

// ---------------------------------------------------------------------------
// Kernel 2: out = relu( ((1/U)*S @ Wv + (N/U)*bv) @ Wo + bo )
// Single block, 8 wave32s. Wave w owns output column tile [16w, 16w+16).
// Both GEMMs use V_WMMA_F32_16X16X4_F32 (full f32 precision, matches ref).
//
// Per ISA 7.12.2 (32-bit A 16x4): lanes 0-15 hold A[M=lane][k0 +0/+1] in the
// two A VGPRs, lanes 16-31 hold A[M=lane-16][k0 +2/+3]. B (4x16) mirrors C/D:
// one K-row per VGPR half, N striped across 16 lanes. C/D: VGPR g holds
// row (g + 8*hi), col = lane&15.
// ---------------------------------------------------------------------------
__global__ __launch_bounds__(256) void epilogue_wmma_kernel(
        const float* __restrict__ S,
        const float* __restrict__ Wv, const float* __restrict__ bv,
        const float* __restrict__ Wo, const float* __restrict__ bo,
        float* __restrict__ out) {
    __shared__ float T[16 * MEM_];           // intermediate [16,128], 8 KB

    const int lane  = threadIdx.x & 31;
    const int w     = threadIdx.x >> 5;      // wave id 0..7 -> column tile
    const int laneM = lane & 15;
    const int hi    = lane >> 4;
    const int col   = w * 16 + laneM;

    // ---- GEMM 1: T = (1/U) * S[16,256] @ Wv[256,128] + (N/U) * bv ----
    v8f acc = {};
#pragma unroll 8
    for (int k0 = 0; k0 < FEAT_; k0 += 4) {
        const int k = k0 + 2 * hi;
        v2f a = *reinterpret_cast<const v2f*>(S + laneM * FEAT_ + k);
        v2f bb;
        bb.x = Wv[(size_t)k * MEM_ + col];
        bb.y = Wv[(size_t)(k + 1) * MEM_ + col];
        acc = __builtin_amdgcn_wmma_f32_16x16x4_f32(
            /*neg_a=*/false, a, /*neg_b=*/false, bb,
            /*c_mod=*/(short)0, acc, /*reuse_a=*/false, /*reuse_b=*/false);
    }
    {
        const float bvv = bv[col];
        const float inv_u = 1.0f / (float)U_;
        const float nb    = (float)N_ / (float)U_;   // 16.0
#pragma unroll
        for (int g = 0; g < 8; ++g) {
            const int row = g + 8 * hi;
            T[row * MEM_ + col] = acc[g] * inv_u + nb * bvv;
        }
    }
    __syncthreads();

    // ---- GEMM 2: out = relu( T[16,128] @ Wo[128,128] + bo ) ----
    v8f acc2 = {};
#pragma unroll 8
    for (int k0 = 0; k0 < MEM_; k0 += 4) {
        const int k = k0 + 2 * hi;
        v2f a = *reinterpret_cast<const v2f*>(&T[laneM * MEM_ + k]);
        v2f bb;
        bb.x = Wo[(size_t)k * MEM_ + col];
        bb.y = Wo[(size_t)(k + 1) * MEM_ + col];
        acc2 = __builtin_amdgcn_wmma_f32_16x16x4_f32(
            false, a, false, bb, (short)0, acc2, false, false);
    }
    {
        const float bov = bo[col];
#pragma unroll
        for (int g = 0; g < 8; ++g) {
            const int row = g + 8 * hi;
            float v = acc2[g] + bov;
            out[row * MEM_ + col] = v > 0.0f ? v : 0.0f;
        }
    }
}

// ---------------------------------------------------------------------------
// Inputs (setup_inputs order): 0:X 1:mem 2:Wk 3:bk 4:Wv 5:bv 6:Wo 7:bo
// mem/Wk/bk are algebraically eliminated (softmax rows sum to 1).
// ---------------------------------------------------------------------------
extern "C" void kernel_launch(void* const* d_in, const int* in_sizes, int n_in,
                              void* d_out, int out_size, void* d_ws, size_t ws_size,
                              hipStream_t stream) {
    const float* X  = (const float*)d_in[0];
    const float* Wv = (const float*)d_in[4];
    const float* bv = (const float*)d_in[5];
    const float* Wo = (const float*)d_in[6];
    const float* bo = (const float*)d_in[7];
    float* out = (float*)d_out;
    float* S   = (float*)d_ws;               // 16*256 floats = 16 KB scratch

    zero_S_kernel<<<(B_ * FEAT_ + 255) / 256, 256, 0, stream>>>(S);
    reduce_X_kernel<<<B_ * 64, 256, 0, stream>>>(X, S);
    epilogue_wmma_kernel<<<1, 256, 0, stream>>>(S, Wv, bv, Wo, bo, out);
}